// LinearDeltaMemoryInline_74929999446552
// MI455X (gfx1250) — compile-verified
//
#include <hip/hip_runtime.h>
#include <math.h>
#include <stdint.h>

// ---- problem constants (match reference) ----
#define NB      8      // batch
#define SEQ     2048   // sequence length
#define DIM     512    // feature dim
#define CT      16     // chunk (time tile) == WMMA tile
#define NCHUNK  (SEQ / CT)
#define RPB     64     // M rows owned per workgroup
#define LDW     516    // padded LDS row stride (floats): 516 % 64 == 4 -> conflict-free frags
#define UES     17     // padded stride for Us/Es (16x16 scratch)
#define EPSN    1e-5f

typedef __attribute__((ext_vector_type(2))) float v2f;
typedef __attribute__((ext_vector_type(8))) float v8f;
typedef __attribute__((ext_vector_type(4))) unsigned int v4u;
typedef __attribute__((ext_vector_type(4))) int v4i;
typedef __attribute__((ext_vector_type(8))) int v8i;

static __device__ __forceinline__ v8f wmma4(v2f a, v2f b, v8f c) {
  // D(16x16,f32) = A(16x4,f32) * B(4x16,f32) + C
  return __builtin_amdgcn_wmma_f32_16x16x4_f32(
      /*neg_a=*/false, a, /*neg_b=*/false, b,
      /*c_mod=*/(short)0, c, /*reuse_a=*/false, /*reuse_b=*/false);
}

// TDM: DMA one 16x512 f32 X chunk into LDS with padded rows (stride LDW floats),
// using iterate mode: tile = 1 row x 512 elems, 16 iterations,
// global += 512 elems/iter, LDS += 516 elems/iter.
static __device__ __forceinline__ void tdm_load_x_chunk(const float* gptr, unsigned ldsByteAddr) {
  const unsigned long long ga = (unsigned long long)(uintptr_t)gptr;
  v4u g0;
  g0[0] = 1u;                                                   // count=1 (valid D#)
  g0[1] = ldsByteAddr;                                          // lds_addr (bytes)
  g0[2] = (unsigned)(ga & 0xFFFFFFFFull);                       // global_addr[31:0]
  g0[3] = (unsigned)((ga >> 32) & 0x1FFFFFFull) | (2u << 30);   // global_addr[56:32] | type=2
  v8i g1;
  g1[0] = (int)((2u << 16) | (1u << 19));  // workgroup_mask=0, data_size=4B, iterate_enable=1
  g1[1] = (int)(512u << 16);               // tensor_dim0 = 512 (lo16 at bits[63:48])
  g1[2] = (int)(16u  << 16);               // tensor_dim1 = 16  (lo16 at bits[95:80])
  g1[3] = (int)(512u << 16);               // tile_dim0   = 512 (bits[127:112])
  g1[4] = 1;                               // tile_dim1 = 1, tile_dim2 = 0
  g1[5] = 512;                             // tensor_dim0_stride = 512 elems
  g1[6] = (int)(512u << 16);               // tensor_dim1_stride lo16
  g1[7] = 0;
  v4i g2;
  g2[0] = 1;                               // tensor_dim2 (unused with iterate)
  g2[1] = LDW;                             // lds_addr_increment per iteration (elems)
  g2[2] = 512;                             // global_addr_increment per iteration (elems)
  g2[3] = (int)(15u << 16);                // iterate_count = 15 -> 16 iterations
  v4i g3 = {0, 0, 0, 0};
#if defined(__clang_major__) && __clang_major__ >= 23
  v8i g4 = {0, 0, 0, 0, 0, 0, 0, 0};
  __builtin_amdgcn_tensor_load_to_lds(g0, g1, g2, g3, g4, 0);
#else
  __builtin_amdgcn_tensor_load_to_lds(g0, g1, g2, g3, 0);
#endif
}

__global__ __launch_bounds__(128, 1)
void delta_memory_chunked(const float* __restrict__ x,
                          const float* __restrict__ Minit,
                          const float* __restrict__ etaRaw,
                          const float* __restrict__ alphaRaw,
                          float* __restrict__ out) {
  extern __shared__ float smem[];
  float* Ms   = smem;                  // [RPB][LDW]    state slice (rows rowBase..+63)
  float* Xs   = Ms + RPB * LDW;        // [2][CT][LDW]  double-buffered x chunk (TDM target)
  float* Ps   = Xs + 2 * CT * LDW;     // [16][16]      P = X X^T
  float* Ss   = Ps + 256;              // [16][16]      S[j][t] = k_j . x_t
  float* Us   = Ss + 256;              // [RPB][UES]    U = M0 X^T
  float* Es   = Us + RPB * UES;        // [RPB][UES]    scaled errors for update GEMM
  float* nInv = Es + RPB * UES;        // [16]          1/max(||x_t||, eps)

  const int tid  = threadIdx.x;
  const int lane = tid & 31;
  const int wave = tid >> 5;       // 4 waves; wave w owns M rows [16w,16w+16)
  const int half = lane >> 4;      // WMMA lane-half
  const int lr   = lane & 15;
  const int b       = blockIdx.y;
  const int rowBase = blockIdx.x * RPB;

  const unsigned xsBase  = (unsigned)(unsigned long long)(uintptr_t)(void*)Xs;
  const unsigned xsBytes = (unsigned)(CT * LDW * sizeof(float));
  const float* xb = x + (size_t)b * SEQ * DIM;

  // kick off chunk 0 DMA immediately (one TDM issue per block, from wave 0)
  if (wave == 0) tdm_load_x_chunk(xb, xsBase);

  const float eta   = 0.5f / (1.0f + __expf(-etaRaw[0]));            // sigmoid * MAX_LR
  const float alpha = 0.5f + 0.5f / (1.0f + __expf(-alphaRaw[0]));   // MIN_DECAY + sig*(1-..)
  float apow[CT + 1];
  apow[0] = 1.0f;
#pragma unroll
  for (int p = 1; p <= CT; ++p) apow[p] = apow[p - 1] * alpha;
  const float a16 = apow[CT];

  // ---- load resident M slice (broadcast M_init) ----
  for (int i = tid; i < RPB * (DIM / 4); i += blockDim.x) {
    const int r = i >> 7, c4 = i & 127;
    float4 v = reinterpret_cast<const float4*>(Minit)[(size_t)(rowBase + r) * (DIM / 4) + c4];
    *reinterpret_cast<float4*>(&Ms[r * LDW + c4 * 4]) = v;
  }
  __syncthreads();

  for (int c = 0; c < NCHUNK; ++c) {
    const int t0 = c * CT;

    // ---- pipeline: issue DMA for chunk c+1, then fence chunk c (in-order TDM) ----
    if (wave == 0) {
      if (c + 1 < NCHUNK) {
        tdm_load_x_chunk(xb + (size_t)(t0 + CT) * DIM, xsBase + ((unsigned)(c + 1) & 1u) * xsBytes);
        __builtin_amdgcn_s_wait_tensorcnt(1);   // chunk c has landed
      } else {
        __builtin_amdgcn_s_wait_tensorcnt(0);
      }
    }
    __syncthreads();
    const float* Xc = Xs + (c & 1) * CT * LDW;

    // ---- WMMA: U = M * X^T (per wave 16x16 tile, K=512); wave0 also P = X X^T ----
    v8f uAcc = {0, 0, 0, 0, 0, 0, 0, 0};
    v8f pAcc = {0, 0, 0, 0, 0, 0, 0, 0};
    const int mrow = (wave * 16 + lr) * LDW;
    const int xrow = lr * LDW;
    if (wave == 0) {
#pragma unroll 8
      for (int kb = 0; kb < DIM; kb += 4) {
        v2f xf, mf;
        xf[0] = Xc[xrow + kb + 2 * half];
        xf[1] = Xc[xrow + kb + 1 + 2 * half];
        mf[0] = Ms[mrow + kb + 2 * half];
        mf[1] = Ms[mrow + kb + 1 + 2 * half];
        uAcc = wmma4(mf, xf, uAcc);
        pAcc = wmma4(xf, xf, pAcc);
      }
    } else {
#pragma unroll 8
      for (int kb = 0; kb < DIM; kb += 4) {
        v2f xf, mf;
        xf[0] = Xc[xrow + kb + 2 * half];
        xf[1] = Xc[xrow + kb + 1 + 2 * half];
        mf[0] = Ms[mrow + kb + 2 * half];
        mf[1] = Ms[mrow + kb + 1 + 2 * half];
        uAcc = wmma4(mf, xf, uAcc);
      }
    }
#pragma unroll
    for (int g = 0; g < 8; ++g)
      Us[(wave * 16 + g + 8 * half) * UES + lr] = uAcc[g];
    if (wave == 0) {
#pragma unroll
      for (int g = 0; g < 8; ++g) Ps[(g + 8 * half) * 16 + lr] = pAcc[g];
    }
    __syncthreads();

    // ---- norms and cross-dot matrix ----
    if (tid < CT) {
      const float nn = sqrtf(Ps[tid * 16 + tid]);   // ||x_t||
      nInv[tid] = 1.0f / fmaxf(nn, EPSN);
    }
    __syncthreads();
    for (int i = tid; i < 256; i += blockDim.x)
      Ss[i] = Ps[i] * nInv[i >> 4];                 // S[j][t] = (x_j.x_t)/n_j
    __syncthreads();

    // ---- tiny in-chunk sequential recurrence (per row, fully unrolled) ----
    if (tid < RPB) {
      const int o = tid;
      float err[CT];
#pragma unroll
      for (int t = 0; t < CT; ++t) {
        float acc = 0.0f;
#pragma unroll
        for (int j = 0; j < CT; ++j)
          if (j < t) acc += apow[t - 1 - j] * err[j] * Ss[j * 16 + t];
        const float ov = apow[t] * Us[o * UES + t] + eta * acc;   // out_t[o]
        out[((size_t)b * SEQ + (t0 + t)) * DIM + (rowBase + o)] = ov;
        err[t] = ov * (1.0f - nInv[t]);                           // out - M k
      }
#pragma unroll
      for (int j = 0; j < CT; ++j)
        Es[o * UES + j] = eta * apow[CT - 1 - j] * err[j];        // pre-scaled for update
    }
    __syncthreads();

    // ---- WMMA: M = alpha^16 * M + E (64x16) * K (16x512), K[j][d]=x_j[d]/n_j ----
    for (int ct2 = 0; ct2 < DIM / 16; ++ct2) {
      const int d0 = ct2 * 16;
      v8f cf;
#pragma unroll
      for (int g = 0; g < 8; ++g)
        cf[g] = a16 * Ms[(wave * 16 + g + 8 * half) * LDW + d0 + lr];
#pragma unroll
      for (int kk = 0; kk < 4; ++kk) {
        const int k0 = kk * 4 + 2 * half;
        v2f ef, kf;
        ef[0] = Es[(wave * 16 + lr) * UES + k0];
        ef[1] = Es[(wave * 16 + lr) * UES + k0 + 1];
        kf[0] = Xc[k0 * LDW + d0 + lr] * nInv[k0];
        kf[1] = Xc[(k0 + 1) * LDW + d0 + lr] * nInv[k0 + 1];
        cf = wmma4(ef, kf, cf);
      }
#pragma unroll
      for (int g = 0; g < 8; ++g)
        Ms[(wave * 16 + g + 8 * half) * LDW + d0 + lr] = cf[g];
    }
    __syncthreads();
  }

  // ---- emit M_final after outs ----
  const size_t OUTM = (size_t)NB * SEQ * DIM;
  for (int i = tid; i < RPB * (DIM / 4); i += blockDim.x) {
    const int r = i >> 7, c4 = i & 127;
    float4 v = *reinterpret_cast<const float4*>(&Ms[r * LDW + c4 * 4]);
    reinterpret_cast<float4*>(out + OUTM)[((size_t)b * DIM + rowBase + r) * (DIM / 4) + c4] = v;
  }
}

extern "C" void kernel_launch(void* const* d_in, const int* in_sizes, int n_in,
                              void* d_out, int out_size, void* d_ws, size_t ws_size,
                              hipStream_t stream) {
  (void)in_sizes; (void)n_in; (void)out_size; (void)d_ws; (void)ws_size;
  const float* x        = (const float*)d_in[0];
  const float* Minit    = (const float*)d_in[1];
  const float* etaRaw   = (const float*)d_in[2];
  const float* alphaRaw = (const float*)d_in[3];
  float* out = (float*)d_out;

  const size_t shmem =
      (size_t)(RPB * LDW + 2 * CT * LDW + 256 + 256 + RPB * UES + RPB * UES + 16) * sizeof(float);
  dim3 grid(DIM / RPB, NB);  // (8 row-blocks, 8 batches) = 64 workgroups
  delta_memory_chunked<<<grid, 128, shmem, stream>>>(x, Minit, etaRaw, alphaRaw, out);
}